// GAT_34110630265039
// MI455X (gfx1250) — compile-verified
//
#include <hip/hip_runtime.h>
#include <hip/hip_bf16.h>
#include <limits.h>

typedef __attribute__((ext_vector_type(2))) float v2f;
typedef __attribute__((ext_vector_type(8))) float v8f;

#define NEG_SLOPE 0.2f

// ---------------------------------------------------------------------------
// GEMM: C[Nrows x Ncols] = A[Nrows x K] @ B[K x Ncols], fp32 WMMA 16x16x4.
// One wave per 16x16 output tile. Nrows, Ncols multiples of 16.
// A layout per lane: row M = tileM*16 + lane%16, VGPR j -> K = k0 + 2*(lane/16) + j
// B layout per lane: col N = tileN*16 + lane%16, VGPR j -> K = k0 + 2*(lane/16) + j
// D layout per lane: VGPR j -> row = tileM*16 + 8*(lane/16) + j, col N
// ---------------------------------------------------------------------------
__global__ void __launch_bounds__(256)
gemm_wmma_f32(const float* __restrict__ A, const float* __restrict__ B,
              float* __restrict__ C, int Nrows, int K, int Ncols) {
    int wave = (int)((blockIdx.x * blockDim.x + threadIdx.x) >> 5);
    int lane = (int)(threadIdx.x & 31);
    int tilesN = Ncols >> 4;
    int tileM  = wave / tilesN;
    int tileN  = wave - tileM * tilesN;
    if (tileM * 16 >= Nrows) return;           // uniform per-wave exit

    int m  = tileM * 16 + (lane & 15);
    int n  = tileN * 16 + (lane & 15);
    int kb = (lane >> 4) * 2;

    v8f acc = {};
    for (int k0 = 0; k0 < K; k0 += 4) {
        const float* ap = A + (size_t)m * K + (k0 + kb);
        v2f a, b;
        a.x = ap[0];
        a.y = ap[1];
        b.x = B[(size_t)(k0 + kb)     * Ncols + n];
        b.y = B[(size_t)(k0 + kb + 1) * Ncols + n];
        acc = __builtin_amdgcn_wmma_f32_16x16x4_f32(
            /*neg_a=*/false, a, /*neg_b=*/false, b,
            /*c_mod=*/(short)0, acc, /*reuse_a=*/false, /*reuse_b=*/false);
    }

    int rbase = tileM * 16 + ((lane >> 4) << 3);
#pragma unroll
    for (int j = 0; j < 8; ++j) {
        C[(size_t)(rbase + j) * Ncols + n] = acc[j];
    }
}

// ---------------------------------------------------------------------------
// Per-node attention coefficients: al_s[n,h] = dot(h[n,h,:], a_src[h,:]) etc.
// ---------------------------------------------------------------------------
__global__ void attn_coef(const float* __restrict__ h,
                          const float* __restrict__ a_src,
                          const float* __restrict__ a_dst,
                          float* __restrict__ al_s, float* __restrict__ al_d,
                          int N, int H, int Cc) {
    int idx = blockIdx.x * blockDim.x + threadIdx.x;   // n*H + head
    if (idx >= N * H) return;
    int n = idx / H, hd = idx - n * H;
    const float* hp = h + (size_t)n * H * Cc + hd * Cc;
    const float* as = a_src + hd * Cc;
    const float* ad = a_dst + hd * Cc;
    float ss = 0.f, sd = 0.f;
    for (int c = 0; c < Cc; ++c) {
        float v = hp[c];
        ss += v * as[c];
        sd += v * ad[c];
    }
    al_s[idx] = ss;
    al_d[idx] = sd;
}

// ---------------------------------------------------------------------------
// Zero out/den, init per-(node,head) running max to INT_MIN (order-preserving
// int-mapped float domain).
// ---------------------------------------------------------------------------
__global__ void init_layer(float* __restrict__ out, float* __restrict__ den,
                           int* __restrict__ mmax, size_t nOut, int nNH) {
    size_t i = (size_t)blockIdx.x * blockDim.x + threadIdx.x;
    size_t stride = (size_t)gridDim.x * blockDim.x;
    for (size_t k = i; k < nOut; k += stride) out[k] = 0.f;
    for (size_t k = i; k < (size_t)nNH; k += stride) {
        den[k]  = 0.f;
        mmax[k] = INT_MIN;
    }
}

__device__ __forceinline__ int ordered_int(float x) {
    int t = __float_as_int(x);
    return (t >= 0) ? t : (t ^ 0x7FFFFFFF);   // involution, order-preserving
}

// ---------------------------------------------------------------------------
// Edge logits + LeakyReLU + segment max (atomicMax on mapped ints).
// Edges e < E come from edge_index; e in [E, E+N) are self loops.
// ---------------------------------------------------------------------------
__global__ void edge_logits(const int* __restrict__ ei,
                            const float* __restrict__ al_s,
                            const float* __restrict__ al_d,
                            float* __restrict__ ebuf, int* __restrict__ mmax,
                            int E, int N, int H) {
    int idx = blockIdx.x * blockDim.x + threadIdx.x;   // e*H + head
    int ET = E + N;
    if (idx >= ET * H) return;
    int e = idx / H, hd = idx - e * H;
    int s, d;
    if (e < E) { s = ei[e]; d = ei[E + e]; } else { s = d = e - E; }
    float x = al_s[s * H + hd] + al_d[d * H + hd];
    x = (x > 0.f) ? x : NEG_SLOPE * x;                 // LeakyReLU
    ebuf[idx] = x;
    atomicMax(&mmax[d * H + hd], ordered_int(x));
}

// ---------------------------------------------------------------------------
// ex = exp(e - m[dst]); segment sum into den[dst]. ebuf overwritten with ex.
// ---------------------------------------------------------------------------
__global__ void edge_exp(const int* __restrict__ ei, float* __restrict__ ebuf,
                         const int* __restrict__ mmax, float* __restrict__ den,
                         int E, int N, int H) {
    int idx = blockIdx.x * blockDim.x + threadIdx.x;
    int ET = E + N;
    if (idx >= ET * H) return;
    int e = idx / H, hd = idx - e * H;
    int d = (e < E) ? ei[E + e] : (e - E);
    int mi = mmax[d * H + hd];
    mi = (mi >= 0) ? mi : (mi ^ 0x7FFFFFFF);           // inverse map
    float m  = __int_as_float(mi);
    float ex = __expf(ebuf[idx] - m);
    ebuf[idx] = ex;
    atomicAdd(&den[d * H + hd], ex);
}

// ---------------------------------------------------------------------------
// Weighted aggregation: out[dst] += h[src] * alpha. One wave per edge; each
// lane owns 8 consecutive channels (D=256 = 32*8), so one head per lane.
// ---------------------------------------------------------------------------
__global__ void __launch_bounds__(256)
edge_aggregate(const int* __restrict__ ei, const float* __restrict__ h,
               const float* __restrict__ ebuf, const float* __restrict__ den,
               float* __restrict__ out, int E, int N, int H, int Cc) {
    int gw   = (int)((blockIdx.x * blockDim.x + threadIdx.x) >> 5);
    int lane = (int)(threadIdx.x & 31);
    int ET = E + N;
    if (gw >= ET) return;
    int s, d;
    if (gw < E) { s = ei[gw]; d = ei[E + gw]; } else { s = d = gw - E; }
    int D  = H * Cc;
    int c0 = lane * (D >> 5);                          // 8 channels per lane
    int hd = c0 / Cc;
    float alpha = ebuf[gw * H + hd] / den[d * H + hd];
    const float4* hp = (const float4*)(h + (size_t)s * D + c0);
    float4 v0 = hp[0], v1 = hp[1];
    float* op = out + (size_t)d * D + c0;
    atomicAdd(op + 0, v0.x * alpha);
    atomicAdd(op + 1, v0.y * alpha);
    atomicAdd(op + 2, v0.z * alpha);
    atomicAdd(op + 3, v0.w * alpha);
    atomicAdd(op + 4, v1.x * alpha);
    atomicAdd(op + 5, v1.y * alpha);
    atomicAdd(op + 6, v1.z * alpha);
    atomicAdd(op + 7, v1.w * alpha);
}

// ---------------------------------------------------------------------------
// In-place: x = elu(x + b) (between layers).
// ---------------------------------------------------------------------------
__global__ void bias_elu(float* __restrict__ buf, const float* __restrict__ b,
                         size_t total, int D) {
    size_t i = (size_t)blockIdx.x * blockDim.x + threadIdx.x;
    size_t stride = (size_t)gridDim.x * blockDim.x;
    for (size_t k = i; k < total; k += stride) {
        float x = buf[k] + b[k % D];
        buf[k] = (x > 0.f) ? x : (__expf(x) - 1.f);
    }
}

// ---------------------------------------------------------------------------
// Column sums of out1 into pooled[D] (atomics). pooled pre-zeroed.
// ---------------------------------------------------------------------------
__global__ void zero_pooled(float* __restrict__ pooled, int D) {
    int t = blockIdx.x * blockDim.x + threadIdx.x;
    if (t < D) pooled[t] = 0.f;
}

__global__ void __launch_bounds__(256)
pool_sum(const float* __restrict__ out1, float* __restrict__ pooled,
         int N, int D) {
    int c = threadIdx.x;                                // one column per thread
    float acc = 0.f;
    for (int r = blockIdx.x; r < N; r += gridDim.x)
        acc += out1[(size_t)r * D + c];
    atomicAdd(&pooled[c], acc);
}

// ---------------------------------------------------------------------------
// Final head: pooled_mean = pooled/N + b1; logits = pooled_mean @ cls_W + cls_b;
// softmax over NC=16 classes. Single wave.
// ---------------------------------------------------------------------------
__global__ void head_softmax(const float* __restrict__ pooled,
                             const float* __restrict__ b1,
                             const float* __restrict__ clsW,
                             const float* __restrict__ clsb,
                             float* __restrict__ outp, int N, int D, int NC) {
    int lane = threadIdx.x;
    float v = 0.f;
    if (lane < NC) {
        float invN = 1.0f / (float)N;
        for (int d0 = 0; d0 < D; ++d0) {
            float p = pooled[d0] * invN + b1[d0];
            v += p * clsW[d0 * NC + lane];
        }
        v += clsb[lane];
    }
    float mx = (lane < NC) ? v : -3.4e38f;
    for (int o = 8; o > 0; o >>= 1) mx = fmaxf(mx, __shfl_xor(mx, o, 16));
    float ex = (lane < NC) ? __expf(v - mx) : 0.f;
    float sum = ex;
    for (int o = 8; o > 0; o >>= 1) sum += __shfl_xor(sum, o, 16);
    if (lane < NC) outp[lane] = ex / sum;
}

// ---------------------------------------------------------------------------
// Host-side orchestration
// ---------------------------------------------------------------------------
extern "C" void kernel_launch(void* const* d_in, const int* in_sizes, int n_in,
                              void* d_out, int out_size, void* d_ws, size_t ws_size,
                              hipStream_t stream) {
    const float* x      = (const float*)d_in[0];
    const int*   ei     = (const int*)  d_in[1];
    const float* W0     = (const float*)d_in[2];
    const float* a0_src = (const float*)d_in[3];
    const float* a0_dst = (const float*)d_in[4];
    const float* b0     = (const float*)d_in[5];
    const float* W1     = (const float*)d_in[6];
    const float* a1_src = (const float*)d_in[7];
    const float* a1_dst = (const float*)d_in[8];
    const float* b1     = (const float*)d_in[9];
    const float* clsW   = (const float*)d_in[10];
    const float* clsb   = (const float*)d_in[11];
    float* outp = (float*)d_out;

    const int D  = in_sizes[5];          // 256
    const int F  = in_sizes[2] / D;      // 64
    const int N  = in_sizes[0] / F;      // 50000
    const int E  = in_sizes[1] / 2;      // 800000
    const int H  = 4;
    const int Cc = D / H;                // 64
    const int NC = in_sizes[11];         // 16
    const int ET = E + N;

    // workspace layout (256B aligned slabs)
    size_t off = 0;
    auto slab = [&](size_t bytes) {
        size_t o = off;
        off = (off + bytes + 255) & ~(size_t)255;
        return o;
    };
    char* ws = (char*)d_ws;
    float* hbuf   = (float*)(ws + slab((size_t)N * D * 4));
    float* obuf   = (float*)(ws + slab((size_t)N * D * 4));
    float* al_s   = (float*)(ws + slab((size_t)N * H * 4));
    float* al_d   = (float*)(ws + slab((size_t)N * H * 4));
    float* ebuf   = (float*)(ws + slab((size_t)ET * H * 4));
    int*   mmax   = (int*)  (ws + slab((size_t)N * H * 4));
    float* den    = (float*)(ws + slab((size_t)N * H * 4));
    float* pooled = (float*)(ws + slab((size_t)D * 4));
    (void)ws_size; (void)n_in; (void)out_size;

    const int tilesPerLayer = (N / 16) * (D / 16);
    const int gemmBlocks    = (tilesPerLayer + 7) / 8;      // 8 waves / block
    const int nhThreads     = N * H;
    const int nhBlocks      = (nhThreads + 255) / 256;
    const int edgThreads    = ET * H;
    const int edgBlocks     = (edgThreads + 255) / 256;
    const int aggBlocks     = (ET + 7) / 8;                 // wave per edge
    const int initBlocks    = 2048;

    // ---------------- Layer 0 ----------------
    gemm_wmma_f32<<<gemmBlocks, 256, 0, stream>>>(x, W0, hbuf, N, F, D);
    attn_coef<<<nhBlocks, 256, 0, stream>>>(hbuf, a0_src, a0_dst, al_s, al_d, N, H, Cc);
    init_layer<<<initBlocks, 256, 0, stream>>>(obuf, den, mmax, (size_t)N * D, N * H);
    edge_logits<<<edgBlocks, 256, 0, stream>>>(ei, al_s, al_d, ebuf, mmax, E, N, H);
    edge_exp<<<edgBlocks, 256, 0, stream>>>(ei, ebuf, mmax, den, E, N, H);
    edge_aggregate<<<aggBlocks, 256, 0, stream>>>(ei, hbuf, ebuf, den, obuf, E, N, H, Cc);
    bias_elu<<<initBlocks, 256, 0, stream>>>(obuf, b0, (size_t)N * D, D);

    // ---------------- Layer 1 ----------------
    gemm_wmma_f32<<<gemmBlocks, 256, 0, stream>>>(obuf, W1, hbuf, N, D, D);
    attn_coef<<<nhBlocks, 256, 0, stream>>>(hbuf, a1_src, a1_dst, al_s, al_d, N, H, Cc);
    init_layer<<<initBlocks, 256, 0, stream>>>(obuf, den, mmax, (size_t)N * D, N * H);
    edge_logits<<<edgBlocks, 256, 0, stream>>>(ei, al_s, al_d, ebuf, mmax, E, N, H);
    edge_exp<<<edgBlocks, 256, 0, stream>>>(ei, ebuf, mmax, den, E, N, H);
    edge_aggregate<<<aggBlocks, 256, 0, stream>>>(ei, hbuf, ebuf, den, obuf, E, N, H, Cc);

    // ---------------- Pool + classifier + softmax ----------------
    zero_pooled<<<1, 256, 0, stream>>>(pooled, D);
    pool_sum<<<512, 256, 0, stream>>>(obuf, pooled, N, D);
    head_softmax<<<1, 32, 0, stream>>>(pooled, b1, clsW, clsb, outp, N, D, NC);
}